// RangeLoss_6150393167991
// MI455X (gfx1250) — compile-verified
//
#include <hip/hip_runtime.h>

// ---------------------------------------------------------------------------
// RangeLoss for MI455X (gfx1250, wave32).
// Streaming memory-bound reduction: 128 MiB in -> 1 float out.
//   per elem: in_range = (p*1.05 > t) && (p*0.95 < t)
//             contrib  = in_range ? 0 : (p-t)^2 + (ln p - ln t)^2
//   result = sum(contrib) / N^2
// Reduction uses V_WMMA_F32_16X16X4_F32 (exact fp32 matrix op) for the
// intra-wave tree, LDS for inter-wave, d_ws for inter-block partials
// (plain stores -> deterministic), and a final 1-block kernel.
// ---------------------------------------------------------------------------

typedef __attribute__((ext_vector_type(2))) float v2f;
typedef __attribute__((ext_vector_type(8))) float v8f;

#define RL_CONF   0.05f
#define RL_LN2    0.69314718055994530942f
#define RL_NBLK   2048
#define RL_NTHR   256

__device__ __forceinline__ float rl_elem(float p, float t) {
    // reference: if (p*(1+c) > t) && (p*(1-c) < t) then t' = p  (contrib 0)
    bool in_range = (p * (1.0f + RL_CONF) > t) && (p * (1.0f - RL_CONF) < t);
    float d  = p - t;
    // ln(x) = log2(x) * ln(2); v_log_f32 is native log2
    float dl = (__builtin_amdgcn_logf(p) - __builtin_amdgcn_logf(t)) * RL_LN2;
    float v  = __builtin_fmaf(d, d, dl * dl);
    return in_range ? 0.0f : v;
}

// Exact fp32 wave reduction via V_WMMA_F32_16X16X4_F32:
//   A (16x4) = the 64 lane accumulators (2 per lane), B (4x16) = ones, C = 0.
//   D[m][n] = sum_k A[m][k]  (row sums, replicated across columns).
//   D layout: lane L<16 vgpr v = D[v][L]; lane L>=16 vgpr v = D[v+8][L-16].
//   sum of 8 D components = low-row total (lanes 0-15) / high-row total
//   (lanes 16-31); one xor-16 shuffle completes the full-wave sum.
// Then wave leaders combine through LDS. Valid result on thread 0 only.
__device__ __forceinline__ float rl_block_reduce(float acc0, float acc1) {
    __shared__ float lds[RL_NTHR / 32];

    v2f a; a[0] = acc0; a[1] = acc1;
    v2f b; b[0] = 1.0f; b[1] = 1.0f;
    v8f c = {};
    v8f d = __builtin_amdgcn_wmma_f32_16x16x4_f32(
        /*neg_a=*/false, a, /*neg_b=*/false, b,
        /*c_mod=*/(short)0, c, /*reuse_a=*/false, /*reuse_b=*/false);

    float s = ((d[0] + d[1]) + (d[2] + d[3])) + ((d[4] + d[5]) + (d[6] + d[7]));
    s += __shfl_xor(s, 16, 32);            // combine low/high row halves

    const int lane = threadIdx.x & 31;
    const int wid  = threadIdx.x >> 5;
    if (lane == 0) lds[wid] = s;
    __syncthreads();

    float blocksum = 0.0f;
    if (threadIdx.x == 0) {
        #pragma unroll
        for (int w = 0; w < RL_NTHR / 32; ++w) blocksum += lds[w];
    }
    return blocksum;
}

__global__ __launch_bounds__(RL_NTHR)
void rangeloss_partial_kernel(const float* __restrict__ preds,
                              const float* __restrict__ target,
                              float* __restrict__ partials,
                              int n4, int n) {
    const float4* __restrict__ p4 = (const float4*)preds;
    const float4* __restrict__ t4 = (const float4*)target;

    float acc0 = 0.0f, acc1 = 0.0f;
    const int stride = gridDim.x * blockDim.x;
    for (int i = blockIdx.x * blockDim.x + threadIdx.x; i < n4; i += stride) {
        float4 p = p4[i];
        float4 t = t4[i];
        acc0 += rl_elem(p.x, t.x) + rl_elem(p.y, t.y);
        acc1 += rl_elem(p.z, t.z) + rl_elem(p.w, t.w);
    }
    // scalar tail (empty for N = 16M; kept for generality)
    if (blockIdx.x == 0 && threadIdx.x == 0) {
        for (int i = n4 * 4; i < n; ++i) acc0 += rl_elem(preds[i], target[i]);
    }
    // reconverged here: EXEC all-1s for the WMMA
    float blocksum = rl_block_reduce(acc0, acc1);
    if (threadIdx.x == 0) partials[blockIdx.x] = blocksum;   // plain store: deterministic
}

__global__ __launch_bounds__(RL_NTHR)
void rangeloss_final_kernel(const float* __restrict__ partials,
                            float* __restrict__ out,
                            int nparts, float scale) {
    float acc0 = 0.0f, acc1 = 0.0f;
    for (int i = threadIdx.x; i < nparts; i += 2 * RL_NTHR) {
        acc0 += partials[i];
        int j = i + RL_NTHR;
        if (j < nparts) acc1 += partials[j];
    }
    float total = rl_block_reduce(acc0, acc1);
    if (threadIdx.x == 0) out[0] = total * scale;
}

extern "C" void kernel_launch(void* const* d_in, const int* in_sizes, int n_in,
                              void* d_out, int out_size, void* d_ws, size_t ws_size,
                              hipStream_t stream) {
    const float* preds  = (const float*)d_in[0];
    const float* target = (const float*)d_in[1];
    float* out      = (float*)d_out;
    float* partials = (float*)d_ws;

    const int n  = in_sizes[0];
    const int n4 = n >> 2;

    int nblk = RL_NBLK;
    if ((size_t)nblk * sizeof(float) > ws_size) {
        nblk = (int)(ws_size / sizeof(float));
        if (nblk < 1) nblk = 1;
    }

    // final scale: reference divides by N and then takes mean over N -> 1/N^2
    const float scale = (float)(1.0 / ((double)n * (double)n));

    rangeloss_partial_kernel<<<nblk, RL_NTHR, 0, stream>>>(preds, target, partials, n4, n);
    rangeloss_final_kernel<<<1, RL_NTHR, 0, stream>>>(partials, out, nblk, scale);
}